// MatrixMPowerSeriesLayer_58377195487395
// MI455X (gfx1250) — compile-verified
//
#include <hip/hip_runtime.h>

typedef __attribute__((ext_vector_type(2))) float v2f;
typedef __attribute__((ext_vector_type(8))) float v8f;

// D = A(16x4 f32) * B(4x16 f32) + C(16x16 f32)
#define WMMA_F32(a, b, c) \
    __builtin_amdgcn_wmma_f32_16x16x4_f32(false, (a), false, (b), (short)0, (c), false, false)

// One wave32 per batch matrix. 32x32 complex matmuls tiled as 2x2 grid of
// 16x16 WMMA tiles, K=32 split into 8 chunks of 4.
__global__ __launch_bounds__(32) void power_series_kernel(
    const float* __restrict__ x,       // [B,2,32,32]
    const float* __restrict__ coeffs,  // [8,2,32,32]
    float* __restrict__ out)           // [B,2,32,32]
{
    __shared__ float lds[2048];  // [0..1023] T_real, [1024..2047] T_imag (row-major 32x32)

    const int lane = threadIdx.x & 31;
    const int half = lane >> 4;   // 0 for lanes 0-15, 1 for 16-31
    const int l15  = lane & 15;
    const int b    = blockIdx.x;

    const float* xr = x + (size_t)b * 2048;
    const float* xi = xr + 1024;
    float* outr = out + (size_t)b * 2048;
    float* outi = outr + 1024;

    // ---- X resident in registers as B-operand fragments ----
    // B frag (4x16): VGPR v holds element [K = kc*4 + 2*half + v][N = j*16 + l15]
    v2f XrB[2][8], XiB[2][8];
    #pragma unroll
    for (int j = 0; j < 2; ++j) {
        #pragma unroll
        for (int kc = 0; kc < 8; ++kc) {
            const int r = kc * 4 + 2 * half;
            const int c = j * 16 + l15;
            XrB[j][kc] = v2f{ xr[r * 32 + c], xr[(r + 1) * 32 + c] };
            XiB[j][kc] = v2f{ xi[r * 32 + c], xi[(r + 1) * 32 + c] };
        }
    }

    // ---- R = C0 (C0 @ I = C0), loaded straight into C/D fragment layout ----
    // C/D frag: VGPR v holds element [M = i*16 + v + 8*half][N = j*16 + l15]
    const float* cr0 = coeffs;
    const float* ci0 = coeffs + 1024;
    v8f Rr[2][2], Ri[2][2];
    #pragma unroll
    for (int i = 0; i < 2; ++i) {
        #pragma unroll
        for (int j = 0; j < 2; ++j) {
            #pragma unroll
            for (int v = 0; v < 8; ++v) {
                const int r = i * 16 + v + 8 * half;
                const int c = j * 16 + l15;
                Rr[i][j][v] = cr0[r * 32 + c];
                Ri[i][j][v] = ci0[r * 32 + c];
            }
        }
    }

    // ---- T = I staged in LDS (imag part = 0) ----
    #pragma unroll
    for (int t = 0; t < 64; ++t) lds[t * 32 + lane] = 0.0f;
    lds[lane * 33] = 1.0f;  // diag of real part; same-wave LDS ops are in-order

    for (int deg = 1; deg < 8; ++deg) {
        // ================= T_new = T @ X =================
        // T as A-operand from LDS, X as B-operand from registers.
        v8f nTr[2][2], nTi[2][2];
        #pragma unroll
        for (int i = 0; i < 2; ++i)
            #pragma unroll
            for (int j = 0; j < 2; ++j) { nTr[i][j] = v8f{}; nTi[i][j] = v8f{}; }

        #pragma unroll
        for (int i = 0; i < 2; ++i) {
            #pragma unroll
            for (int kc = 0; kc < 8; ++kc) {
                // A frag (16x4): VGPR v = element [M = i*16 + l15][K = kc*4 + 2*half + v]
                const float* pr = &lds[(i * 16 + l15) * 32 + kc * 4 + 2 * half];
                const float* pi = pr + 1024;
                v2f ar  = v2f{ pr[0], pr[1] };      // ds_load_b64
                v2f ai  = v2f{ pi[0], pi[1] };
                v2f nai = v2f{ -ai.x, -ai.y };      // f32 WMMA has no A/B neg modifier
                #pragma unroll
                for (int j = 0; j < 2; ++j) {
                    nTr[i][j] = WMMA_F32(ar,  XrB[j][kc], nTr[i][j]);  // +Tr*Xr
                    nTr[i][j] = WMMA_F32(nai, XiB[j][kc], nTr[i][j]);  // -Ti*Xi
                    nTi[i][j] = WMMA_F32(ar,  XiB[j][kc], nTi[i][j]);  // +Tr*Xi
                    nTi[i][j] = WMMA_F32(ai,  XrB[j][kc], nTi[i][j]);  // +Ti*Xr
                }
            }
        }

        // Store T_new back to LDS (old T fully read already; same-wave LDS is in-order).
        #pragma unroll
        for (int i = 0; i < 2; ++i) {
            #pragma unroll
            for (int j = 0; j < 2; ++j) {
                #pragma unroll
                for (int v = 0; v < 8; ++v) {
                    const int r = i * 16 + v + 8 * half;
                    const int c = j * 16 + l15;
                    lds[r * 32 + c]        = nTr[i][j][v];
                    lds[1024 + r * 32 + c] = nTi[i][j][v];
                }
            }
        }

        // ================= R += C_deg @ T_new =================
        // C as A-operand from global (L2-hot 64KB), T_new as B-operand from LDS.
        const float* cr = coeffs + deg * 2048;
        const float* ci = cr + 1024;
        #pragma unroll
        for (int kc = 0; kc < 8; ++kc) {
            v2f TbR[2], TbI[2];
            #pragma unroll
            for (int j = 0; j < 2; ++j) {
                const int r = kc * 4 + 2 * half;
                const int c = j * 16 + l15;
                TbR[j] = v2f{ lds[r * 32 + c],        lds[(r + 1) * 32 + c] };
                TbI[j] = v2f{ lds[1024 + r * 32 + c], lds[1024 + (r + 1) * 32 + c] };
            }
            #pragma unroll
            for (int i = 0; i < 2; ++i) {
                const float* par = &cr[(i * 16 + l15) * 32 + kc * 4 + 2 * half];
                const float* pai = &ci[(i * 16 + l15) * 32 + kc * 4 + 2 * half];
                v2f car  = v2f{ par[0], par[1] };   // global_load_b64
                v2f cai  = v2f{ pai[0], pai[1] };
                v2f ncai = v2f{ -cai.x, -cai.y };
                #pragma unroll
                for (int j = 0; j < 2; ++j) {
                    Rr[i][j] = WMMA_F32(car,  TbR[j], Rr[i][j]);  // +Cr*Tr
                    Rr[i][j] = WMMA_F32(ncai, TbI[j], Rr[i][j]);  // -Ci*Ti
                    Ri[i][j] = WMMA_F32(car,  TbI[j], Ri[i][j]);  // +Cr*Ti
                    Ri[i][j] = WMMA_F32(cai,  TbR[j], Ri[i][j]);  // +Ci*Tr
                }
            }
        }
    }

    // ---- write result ----
    #pragma unroll
    for (int i = 0; i < 2; ++i) {
        #pragma unroll
        for (int j = 0; j < 2; ++j) {
            #pragma unroll
            for (int v = 0; v < 8; ++v) {
                const int r = i * 16 + v + 8 * half;
                const int c = j * 16 + l15;
                outr[r * 32 + c] = Rr[i][j][v];
                outi[r * 32 + c] = Ri[i][j][v];
            }
        }
    }
}

extern "C" void kernel_launch(void* const* d_in, const int* in_sizes, int n_in,
                              void* d_out, int out_size, void* d_ws, size_t ws_size,
                              hipStream_t stream) {
    const float* x      = (const float*)d_in[0];  // [B,2,32,32] f32
    const float* coeffs = (const float*)d_in[1];  // [8,2,32,32] f32
    float* out          = (float*)d_out;          // [B,2,32,32] f32

    const int B = in_sizes[0] / (2 * 32 * 32);    // 16384
    power_series_kernel<<<B, 32, 0, stream>>>(x, coeffs, out);
}